// FNO_38912403701913
// MI455X (gfx1250) — compile-verified
//
#include <hip/hip_runtime.h>
#include <hip/hip_bf16.h>

#define NGRID 4096
#define DV    128
#define KM    32
#define NB    64
#define NL    4

typedef __attribute__((ext_vector_type(2))) float v2f;
typedef __attribute__((ext_vector_type(8))) float v8f;
typedef __attribute__((ext_vector_type(4))) unsigned int ui4;
typedef __attribute__((ext_vector_type(8))) int i8v;
typedef __attribute__((ext_vector_type(4))) int i4v;

#ifndef __has_builtin
#define __has_builtin(x) 0
#endif
#ifndef __has_include
#define __has_include(x) 0
#endif

#if __has_builtin(__builtin_amdgcn_tensor_load_to_lds) && \
    __has_builtin(__builtin_amdgcn_s_wait_tensorcnt)
#define USE_TDM 1
#else
#define USE_TDM 0
#endif

#if __has_include(<hip/amd_detail/amd_gfx1250_TDM.h>)
#define TDM_ARITY6 1
#else
#define TDM_ARITY6 0
#endif

__device__ __forceinline__ v8f wmma4(v2f a, v2f b, v8f c) {
    // V_WMMA_F32_16X16X4_F32: D = A(16x4) * B(4x16) + C(16x16)
    return __builtin_amdgcn_wmma_f32_16x16x4_f32(
        false, a, false, b, (short)0, c, false, false);
}

// ---------------------------------------------------------------------------
// prep: transpose spectral kernel (complex, [L][K][i][j]) into Ktr/Kti [L][K][j][i]
//       and W [L][i][j] into Wt [L][j][i] for coalesced B-fragment loads.
// ---------------------------------------------------------------------------
__global__ void FNO_prep_kernel(const float2* __restrict__ ker,
                                const float* __restrict__ W,
                                float* __restrict__ Ktr, float* __restrict__ Kti,
                                float* __restrict__ Wt) {
    size_t idx = (size_t)blockIdx.x * blockDim.x + threadIdx.x;
    const size_t total = (size_t)NL * KM * DV * DV;
    if (idx < total) {
        size_t j  = idx % DV;
        size_t i  = (idx / DV) % DV;
        size_t lk = idx / ((size_t)DV * DV);
        float2 c  = ker[idx];
        size_t dst = lk * DV * DV + j * DV + i;
        Ktr[dst] = c.x;
        Kti[dst] = c.y;
    }
    if (idx < (size_t)NL * DV * DV) {
        size_t j = idx % DV;
        size_t i = (idx / DV) % DV;
        size_t l = idx / ((size_t)DV * DV);
        Wt[(l * DV + j) * DV + i] = W[idx];
    }
}

// ---------------------------------------------------------------------------
// lift: v[b,n,i] = u[b,n]*lift_w[i] + lift_b[i]
// ---------------------------------------------------------------------------
__global__ void FNO_lift_kernel(const float* __restrict__ u,
                                const float* __restrict__ lw,
                                const float* __restrict__ lb,
                                float* __restrict__ v) {
    size_t idx = (size_t)blockIdx.x * blockDim.x + threadIdx.x;
    const size_t total = (size_t)NB * NGRID * DV;
    if (idx >= total) return;
    size_t i  = idx & (DV - 1);
    size_t bn = idx >> 7;
    v[idx] = u[bn] * lw[i] + lb[i];
}

__global__ void FNO_zero_kernel(float* __restrict__ p, int n) {
    int i = blockIdx.x * blockDim.x + threadIdx.x;
    if (i < n) p[i] = 0.f;
}

// ---------------------------------------------------------------------------
// dft: vh[b][k][i] (rows 0..31 real, 32..63 imag) = sum_n e^{-2pi i k n/N} v[b,n,i]
// Grid: (NB, 8 dv-tiles); block 256 = 8 waves; wave w handles K-chunk of 512 n.
// LDS table stores (cos, -sin) so A-fragments need zero VALU fixup.
// ---------------------------------------------------------------------------
__global__ __launch_bounds__(256)
void FNO_dft_kernel(const float* __restrict__ v, float* __restrict__ vh) {
    __shared__ float2 tw[NGRID];                     // 32 KB: (cos, -sin)
    const int b  = blockIdx.x;
    const int jt = blockIdx.y;                       // dv tile 0..7
    const int tid = threadIdx.x;
    const float w0 = 6.2831853071795864769f / (float)NGRID;
    for (int p = tid; p < NGRID; p += 256) {
        float s, c;
        __sincosf(w0 * (float)p, &s, &c);
        tw[p] = make_float2(c, -s);
    }
    __syncthreads();

    const int wave = tid >> 5;
    const int lane = tid & 31;
    const int m    = lane & 15;                      // A-row / B-col within tile
    const int hi   = lane >> 4;
    const int kb   = hi << 1;                        // K offset 0 or 2
    const int col  = jt * 16 + m;                    // dv column
    const float* vb = v + (size_t)b * NGRID * DV;

    v8f accr0 = {}, acci0 = {}, accr1 = {}, acci1 = {};

    int n0 = wave * 512;
    for (int it = 0; it < 128; ++it, n0 += 4) {
        // B fragment: v rows n0+kb, n0+kb+1 ; column = col
        const float* vp = vb + (size_t)(n0 + kb) * DV + col;
        v2f bf = { vp[0], vp[DV] };
        // A fragments: modes m (tile0) and m+16 (tile1); grid cols n0+kb(+1)
        int p00 = (m * (n0 + kb))            & (NGRID - 1);
        int p01 = (m * (n0 + kb + 1))        & (NGRID - 1);
        int p10 = ((m + 16) * (n0 + kb))     & (NGRID - 1);
        int p11 = ((m + 16) * (n0 + kb + 1)) & (NGRID - 1);
        float2 t00 = tw[p00], t01 = tw[p01], t10 = tw[p10], t11 = tw[p11];
        v2f ar0 = { t00.x, t01.x };
        v2f ai0 = { t00.y, t01.y };                  // already -sin
        v2f ar1 = { t10.x, t11.x };
        v2f ai1 = { t10.y, t11.y };
        accr0 = wmma4(ar0, bf, accr0);
        acci0 = wmma4(ai0, bf, acci0);
        accr1 = wmma4(ar1, bf, accr1);
        acci1 = wmma4(ai1, bf, acci1);
    }

    // K-split across 8 waves -> accumulate with f32 atomics (vh zeroed first)
    float* whb = vh + (size_t)b * (2 * KM * DV);
    #pragma unroll
    for (int g = 0; g < 8; ++g) {
        int mode0 = g + 8 * hi;                      // 0..15
        atomicAdd(&whb[(size_t)mode0 * DV + col],        accr0[g]);
        atomicAdd(&whb[(size_t)(mode0 + KM) * DV + col], acci0[g]);
        int mode1 = mode0 + 16;                      // 16..31
        atomicAdd(&whb[(size_t)mode1 * DV + col],        accr1[g]);
        atomicAdd(&whb[(size_t)(mode1 + KM) * DV + col], acci1[g]);
    }
}

// ---------------------------------------------------------------------------
// spectral: wh[b,k,i] = sum_j kernel[k,i,j] * vh[b,k,j]   (complex)
// out_r = vr*Kr - vi*Ki ; out_i = vr*Ki + vi*Kr  -> 4 real WMMAs per K-step.
// Grid: 32 modes * 4 batch-tiles; block 256 = 8 waves = 8 i-tiles.
// ---------------------------------------------------------------------------
__global__ __launch_bounds__(256)
void FNO_spectral_kernel(const float* __restrict__ vh,
                         const float* __restrict__ Ktr,
                         const float* __restrict__ Kti,
                         float* __restrict__ wh) {
    const int k    = blockIdx.x & 31;
    const int bt   = blockIdx.x >> 5;                // batch tile 0..3
    const int wave = threadIdx.x >> 5;
    const int lane = threadIdx.x & 31;
    const int m    = lane & 15;
    const int hi   = lane >> 4;
    const int kb   = hi << 1;
    const int i0   = wave * 16;

    const float* Brp = Ktr + (size_t)k * DV * DV;    // [j][i]
    const float* Bip = Kti + (size_t)k * DV * DV;
    const int brow = bt * 16 + m;                    // A row = batch index
    const float* Arb = vh + (size_t)brow * (2 * KM * DV) + (size_t)k * DV;
    const float* Aib = Arb + (size_t)KM * DV;

    v8f accr = {}, acci = {};
    for (int j = 0; j < DV; j += 4) {
        int ja = j + kb;
        v2f ar  = { Arb[ja],  Arb[ja + 1] };
        v2f ai  = { Aib[ja],  Aib[ja + 1] };
        v2f ain = { -ai.x, -ai.y };
        const float* br_p = Brp + (size_t)ja * DV + i0 + m;
        const float* bi_p = Bip + (size_t)ja * DV + i0 + m;
        v2f br = { br_p[0], br_p[DV] };
        v2f bi = { bi_p[0], bi_p[DV] };
        accr = wmma4(ar,  br, accr);
        accr = wmma4(ain, bi, accr);
        acci = wmma4(ar,  bi, acci);
        acci = wmma4(ai,  br, acci);
    }

    #pragma unroll
    for (int g = 0; g < 8; ++g) {
        int b = bt * 16 + g + 8 * hi;
        float* o = wh + (size_t)b * (2 * KM * DV);
        o[(size_t)k * DV + i0 + m]        = accr[g];
        o[(size_t)(k + KM) * DV + i0 + m] = acci[g];
    }
}

// ---------------------------------------------------------------------------
// update (fused): vout = ReLU( vin @ W^T  +  E(n,0:64) @ H[b]  +  bias )
// E is the 64-wide irfft basis: col 0 -> 1/N, cols 1..31 -> 2cos/N,
// col 32 -> 0 (imag of mode 0 dropped by irfft), cols 33..63 -> -2sin/N.
// Grid: B*N/16 row-tile blocks; 8 waves = 8 i-tiles. A rows staged in LDS
// with pad-132 row stride (132 % 64 = 4) to avoid bank conflicts.
// The 16x128 f32 tile is staged by the Tensor Data Mover: a single 2D D#
// with pad_enable (interval=128 DW, amount=4 DW) reproduces the 132-float
// row stride in hardware; wave 0 issues it and waits on TENSORcnt.
// ---------------------------------------------------------------------------
__global__ __launch_bounds__(256)
void FNO_update_kernel(const float* __restrict__ vin,
                       const float* __restrict__ wh,
                       const float* __restrict__ Wt,
                       const float* __restrict__ bias,
                       float* __restrict__ vout) {
    __shared__ __align__(16) float lv[16 * 132];     // 8.25 KB
    const int b  = blockIdx.x >> 8;                  // /256
    const int n0 = (blockIdx.x & 255) * 16;
    const int wave = threadIdx.x >> 5;
    const int lane = threadIdx.x & 31;
    const int m    = lane & 15;
    const int hi   = lane >> 4;
    const int kb   = hi << 1;
    const int i0   = wave * 16;

#if USE_TDM
    if (threadIdx.x < 32) {                          // one wave drives the TDM
        unsigned int lds_addr = (unsigned int)(size_t)&lv[0];
        unsigned long long ga =
            (unsigned long long)(size_t)(vin + ((size_t)b * NGRID + n0) * DV);
        // D# group 0: count=1, lds_addr, global_addr[56:0], type=2
        ui4 g0;
        g0[0] = 1u;
        g0[1] = lds_addr;
        g0[2] = (unsigned int)(ga & 0xFFFFFFFFu);
        g0[3] = (unsigned int)((ga >> 32) & 0x01FFFFFFu) | (2u << 30);
        // D# group 1: data_size=4B(2), pad_enable=1, pad_interval=6 (128 DW),
        // pad_amount=3 (4 DW); tensor 128 x NGRID, tile 128 x 16, stride0=128.
        i8v g1;
        g1[0] = (2 << 16) | (1 << 20) | (6 << 22) | (3 << 25);
        g1[1] = (DV & 0xFFFF) << 16;                 // tensor_dim0 = 128
        g1[2] = (NGRID & 0xFFFF) << 16;              // tensor_dim1 = 4096
        g1[3] = (DV & 0xFFFF) << 16;                 // tile_dim0 = 128
        g1[4] = 16;                                  // tile_dim1 = 16
        g1[5] = DV;                                  // tensor_dim0_stride = 128
        g1[6] = 0;
        g1[7] = 0;
        i4v z = { 0, 0, 0, 0 };                      // 2D: groups 2/3 unused
#if TDM_ARITY6
        i8v z8 = {};
        __builtin_amdgcn_tensor_load_to_lds(g0, g1, z, z, z8, 0);
#else
        __builtin_amdgcn_tensor_load_to_lds(g0, g1, z, z, 0);
#endif
        __builtin_amdgcn_s_wait_tensorcnt(0);
    }
    __syncthreads();
#else
    for (int q = threadIdx.x; q < 16 * DV; q += 256) {
        int r = q >> 7, c = q & 127;
        lv[r * 132 + c] = vin[((size_t)b * NGRID + n0 + r) * DV + c];
    }
    __syncthreads();
#endif

    v8f acc = {};
    // spatial linear part: A = v rows (LDS), B = Wt[j][i]
    for (int j = 0; j < DV; j += 4) {
        v2f a = { lv[m * 132 + j + kb], lv[m * 132 + j + kb + 1] };
        const float* wp = Wt + (size_t)(j + kb) * DV + i0 + m;
        v2f bf = { wp[0], wp[DV] };
        acc = wmma4(a, bf, acc);
    }

    // spectral residual (irfft of 32 modes) as E(16x64) @ H(64x16)
    const float* whb = wh + (size_t)b * (2 * KM * DV);
    const float w0   = 6.2831853071795864769f / (float)NGRID;
    const float invN = 1.0f / (float)NGRID;
    const int   n    = n0 + m;                       // this lane's grid row
    for (int mm = 0; mm < 2 * KM; mm += 4) {
        v2f e;
        #pragma unroll
        for (int g = 0; g < 2; ++g) {
            int c = mm + kb + g;
            float val;
            if (c == 0) {
                val = invN;
            } else if (c < KM) {
                int p = (c * n) & (NGRID - 1);
                val = 2.f * invN * __cosf(w0 * (float)p);
            } else if (c == KM) {
                val = 0.f;                           // irfft drops Im(mode 0)
            } else {
                int p = ((c - KM) * n) & (NGRID - 1);
                val = -2.f * invN * __sinf(w0 * (float)p);
            }
            e[g] = val;
        }
        const float* hp = whb + (size_t)(mm + kb) * DV + i0 + m;
        v2f h = { hp[0], hp[DV] };
        acc = wmma4(e, h, acc);
    }

    // epilogue: + bias, ReLU, store
    #pragma unroll
    for (int g = 0; g < 8; ++g) {
        int row = g + 8 * hi;
        int nn  = n0 + row;
        int col = i0 + m;
        float r = acc[g] + bias[(size_t)nn * DV + col];
        vout[((size_t)b * NGRID + nn) * DV + col] = fmaxf(r, 0.f);
    }
}

// ---------------------------------------------------------------------------
// proj: out[b,n] = dot(v[b,n,:], proj_w) + proj_b
// ---------------------------------------------------------------------------
__global__ __launch_bounds__(256)
void FNO_proj_kernel(const float* __restrict__ v,
                     const float* __restrict__ pw,
                     const float* __restrict__ pb,
                     float* __restrict__ out) {
    __shared__ float w[DV];
    if (threadIdx.x < DV) w[threadIdx.x] = pw[threadIdx.x];
    __syncthreads();
    size_t row = (size_t)blockIdx.x * blockDim.x + threadIdx.x;
    if (row >= (size_t)NB * NGRID) return;
    const float* vp = v + row * DV;
    float s = 0.f;
    #pragma unroll 16
    for (int j = 0; j < DV; ++j) s += vp[j] * w[j];
    out[row] = s + pb[0];
}

// ---------------------------------------------------------------------------
extern "C" void kernel_launch(void* const* d_in, const int* in_sizes, int n_in,
                              void* d_out, int out_size, void* d_ws, size_t ws_size,
                              hipStream_t stream) {
    (void)in_sizes; (void)n_in; (void)out_size; (void)ws_size;
    const float*  u      = (const float*)d_in[0];
    const float*  lift_w = (const float*)d_in[1];
    const float*  lift_b = (const float*)d_in[2];
    const float*  proj_w = (const float*)d_in[3];
    const float*  proj_b = (const float*)d_in[4];
    const float2* ker    = (const float2*)d_in[5];
    const float*  W      = (const float*)d_in[6];
    const float*  bias   = (const float*)d_in[7];
    float* out = (float*)d_out;

    char* ws = (char*)d_ws;
    size_t off = 0;
    auto alloc = [&](size_t bytes) -> void* {
        void* p = ws + off;
        off += (bytes + 255) & ~(size_t)255;
        return p;
    };
    float* v0  = (float*)alloc((size_t)NB * NGRID * DV * 4);   // 128 MB
    float* v1  = (float*)alloc((size_t)NB * NGRID * DV * 4);   // 128 MB
    float* vh  = (float*)alloc((size_t)NB * 2 * KM * DV * 4);  // 2 MB
    float* wh  = (float*)alloc((size_t)NB * 2 * KM * DV * 4);  // 2 MB
    float* Ktr = (float*)alloc((size_t)NL * KM * DV * DV * 4); // 8 MB
    float* Kti = (float*)alloc((size_t)NL * KM * DV * DV * 4); // 8 MB
    float* Wt  = (float*)alloc((size_t)NL * DV * DV * 4);      // 256 KB

    {   // weight prep (transposes)
        size_t total = (size_t)NL * KM * DV * DV;
        FNO_prep_kernel<<<dim3((unsigned)((total + 255) / 256)), 256, 0, stream>>>(
            ker, W, Ktr, Kti, Wt);
    }
    {   // lift
        size_t total = (size_t)NB * NGRID * DV;
        FNO_lift_kernel<<<dim3((unsigned)((total + 255) / 256)), 256, 0, stream>>>(
            u, lift_w, lift_b, v0);
    }

    float* vin = v0;
    float* vout = v1;
    const int vhn = NB * 2 * KM * DV;
    for (int l = 0; l < NL; ++l) {
        FNO_zero_kernel<<<dim3((vhn + 255) / 256), 256, 0, stream>>>(vh, vhn);
        FNO_dft_kernel<<<dim3(NB, 8), 256, 0, stream>>>(vin, vh);
        FNO_spectral_kernel<<<dim3(KM * (NB / 16)), 256, 0, stream>>>(
            vh, Ktr + (size_t)l * KM * DV * DV, Kti + (size_t)l * KM * DV * DV, wh);
        FNO_update_kernel<<<dim3(NB * (NGRID / 16)), 256, 0, stream>>>(
            vin, wh, Wt + (size_t)l * DV * DV, bias + (size_t)l * NGRID * DV, vout);
        float* t = vin; vin = vout; vout = t;
    }

    FNO_proj_kernel<<<dim3((NB * NGRID + 255) / 256), 256, 0, stream>>>(
        vin, proj_w, proj_b, out);
}